// Neuron_83889301226253
// MI455X (gfx1250) — compile-verified
//
#include <hip/hip_runtime.h>
#include <math.h>

typedef __attribute__((ext_vector_type(2))) float v2f;
typedef __attribute__((ext_vector_type(4))) float v4f;
typedef __attribute__((ext_vector_type(8))) float v8f;

#define B_ROWS 1024
#define D_COLS 32768
#define WAVES_PER_BLOCK 8
#define THREADS (WAVES_PER_BLOCK * 32)

// One block per batch row. Each wave streams a contiguous 4096-element slice of
// the row with coalesced 128-bit loads, forms masked products in VALU, and uses
// V_WMMA_F32_16X16X4_F32 with an all-ones B matrix as a 64-element reduction
// engine (D[m][n] = sum_k A[m][k] + C[m][n] for every column n).
__global__ __launch_bounds__(THREADS)
void neuron_masked_gemv_kernel(const float* __restrict__ x,
                               const float* __restrict__ w,
                               const float* __restrict__ thr,
                               const float* __restrict__ ru,
                               const float* __restrict__ eps,
                               float* __restrict__ out) {
  const int row  = blockIdx.x;
  const int tid  = threadIdx.x;
  const int wave = tid >> 5;
  const int lane = tid & 31;

  const size_t row_base = (size_t)row * (size_t)D_COLS;
  const int slice = D_COLS / WAVES_PER_BLOCK;   // 4096 elements per wave
  const int base  = wave * slice;

  const v4f* __restrict__ xp = reinterpret_cast<const v4f*>(x + row_base + base);
  const v4f* __restrict__ up = reinterpret_cast<const v4f*>(ru + row_base + base);
  const v4f* __restrict__ wp = reinterpret_cast<const v4f*>(w + base);

  v8f acc = {};                 // WMMA C/D accumulator (16x16 f32 tile)
  v2f bones;                    // all-ones B matrix (4x16): every D column = row sums of A
  bones[0] = 1.0f; bones[1] = 1.0f;

  const int iters = slice / 128;    // 32 iterations; 128 elements per wave-iter
  int idx = lane;                   // float4 index within the slice
  for (int it = 0; it < iters; ++it, idx += 32) {
    // Streaming data: non-temporal so the resident w vector stays in L2.
    v4f xv = __builtin_nontemporal_load(xp + idx);
    v4f uv = __builtin_nontemporal_load(up + idx);
    v4f wv = wp[idx];               // shared across all rows -> L2 hit

    v2f a0, a1;
    a0[0] = (uv.x < 0.9f) ? xv.x * wv.x : 0.0f;
    a0[1] = (uv.y < 0.9f) ? xv.y * wv.y : 0.0f;
    a1[0] = (uv.z < 0.9f) ? xv.z * wv.z : 0.0f;
    a1[1] = (uv.w < 0.9f) ? xv.w * wv.w : 0.0f;

    // D = A x ones + C : matrix unit performs the 64-element reduction.
    acc = __builtin_amdgcn_wmma_f32_16x16x4_f32(false, a0, false, bones,
                                                (short)0, acc, false, false);
    acc = __builtin_amdgcn_wmma_f32_16x16x4_f32(false, a1, false, bones,
                                                (short)0, acc, false, false);
  }

  // Column 0 of the tile: lanes 0-15 hold M=0..7 across the 8 VGPRs,
  // lanes 16-31 hold M=8..15. Sum VGPRs, then combine lane pairs (xor 16).
  float s = acc[0] + acc[1] + acc[2] + acc[3] + acc[4] + acc[5] + acc[6] + acc[7];
  s += __shfl_xor(s, 16, 32);

  __shared__ float wave_sums[WAVES_PER_BLOCK];
  if (lane == 0) wave_sums[wave] = s;
  __syncthreads();

  if (tid == 0) {
    float total = 0.0f;
#pragma unroll
    for (int i = 0; i < WAVES_PER_BLOCK; ++i) total += wave_sums[i];
    // fatigue = 0.9^row  (log2(0.9) = -0.15200309...)
    float fatigue = exp2f((float)row * -0.152003093445262f);
    float ws_v = total * fatigue;
    float noisy_thr = thr[0] + eps[row] * 1e-5f;
    out[row] = (ws_v > noisy_thr) ? tanhf(ws_v) : 0.0f;
  }
}

extern "C" void kernel_launch(void* const* d_in, const int* in_sizes, int n_in,
                              void* d_out, int out_size, void* d_ws, size_t ws_size,
                              hipStream_t stream) {
  const float* x   = (const float*)d_in[0];  // [B, D]
  const float* w   = (const float*)d_in[1];  // [D]
  const float* thr = (const float*)d_in[2];  // [1]
  const float* ru  = (const float*)d_in[3];  // [B, D]
  const float* eps = (const float*)d_in[4];  // [B]
  float* out = (float*)d_out;                // [B]

  neuron_masked_gemv_kernel<<<B_ROWS, THREADS, 0, stream>>>(x, w, thr, ru, eps, out);
}